// EnhancedLocalAttention_38835094291127
// MI455X (gfx1250) — compile-verified
//
#include <hip/hip_runtime.h>
#include <hip/hip_bf16.h>

typedef float v2f __attribute__((ext_vector_type(2)));
typedef float v8f __attribute__((ext_vector_type(8)));

#define EMBED   1024
#define QKV_N   3072
#define HEADS   16
#define HEAD_D  64
#define WIN     256
#define NW_USED 16              // only windows 0..15 reach the sliced output
#define ROWS_Q  2176            // 15*128 + 256 : last seq row needed per batch
#define SEQ     4096
#define BATCH   4

// ---------------------------------------------------------------------------
// fp32 WMMA GEMM + bias:  C[b] = A[b](MxK, lda=K) * B(KxN, ldb=N) + bias
// Register-blocked: each wave owns a 32x64 C tile = 2(M) x 4(N) 16x16 tiles,
// 8 accumulators. Per K-step of 4: 2 A b64 loads + 8 B b32 loads -> 8 wmma
// (V_WMMA_F32_16X16X4_F32), ~10.7 flops/byte from cache.
// Block = 8 waves as 2(M) x 4(N) -> 64x256 block tile.
// A fragment (16x4 f32): lane L<16 : {A[r][k],A[r][k+1]}, L>=16 : {A[r][k+2],A[r][k+3]}
// B fragment (4x16 f32): symmetric with N on lanes.
// C/D (v8f): lane r holds rows m0+half*8 .. +7 of column col0+r.
// ---------------------------------------------------------------------------
__global__ __launch_bounds__(256)
void gemm_bias_wmma(const float* __restrict__ A, const float* __restrict__ Bm,
                    const float* __restrict__ bias, float* __restrict__ C,
                    int K, int N, long long strideA, long long strideC)
{
    const int lane  = threadIdx.x & 31;
    const int wave  = threadIdx.x >> 5;
    const int r     = lane & 15;
    const int half  = lane >> 4;
    const int waveN = wave & 3;      // 0..3
    const int waveM = wave >> 2;     // 0..1

    const int col0 = blockIdx.x * 256 + waveN * 64;   // wave cols col0..col0+63
    const int row0 = blockIdx.y * 64  + waveM * 32;   // wave rows row0..row0+31
    const int b    = blockIdx.z;

    const float* Ab = A + (long long)b * strideA;
    float*       Cb = C + (long long)b * strideC;

    v8f acc[2][4];
#pragma unroll
    for (int m = 0; m < 2; ++m)
#pragma unroll
        for (int c = 0; c < 4; ++c) acc[m][c] = {};

    const float* aptr0 = Ab + (long long)(row0 + r) * K + 2 * half;
    const float* aptr1 = aptr0 + (long long)16 * K;
    const float* bptr  = Bm + (long long)(2 * half) * N + col0 + r;

#pragma unroll 2
    for (int k = 0; k < K; k += 4) {
        v2f a0 = *(const v2f*)aptr0;      // A rows row0+r    , K-contiguous
        v2f a1 = *(const v2f*)aptr1;      // A rows row0+16+r
        v2f bb[4];
#pragma unroll
        for (int c = 0; c < 4; ++c) {
            bb[c].x = bptr[c * 16];
            bb[c].y = bptr[c * 16 + N];
        }
#pragma unroll
        for (int c = 0; c < 4; ++c) {
            acc[0][c] = __builtin_amdgcn_wmma_f32_16x16x4_f32(
                false, a0, false, bb[c], (short)0, acc[0][c], false, false);
            acc[1][c] = __builtin_amdgcn_wmma_f32_16x16x4_f32(
                false, a1, false, bb[c], (short)0, acc[1][c], false, false);
        }
        aptr0 += 4;
        aptr1 += 4;
        bptr  += 4 * (long long)N;
    }

#pragma unroll
    for (int c = 0; c < 4; ++c) {
        const int col = col0 + c * 16 + r;
        const float bv = bias[col];
#pragma unroll
        for (int m = 0; m < 2; ++m) {
            const int m0 = row0 + m * 16 + half * 8;
#pragma unroll
            for (int i = 0; i < 8; ++i)
                Cb[(long long)(m0 + i) * N + col] = acc[m][c][i] + bv;
        }
    }
}

// ---------------------------------------------------------------------------
// Windowed "attention" per (window n, head h, batch b).
//   Q,K,V : 64x256 with Q[d][w] = qkv[b][n*128+w][sel*1024 + h*64 + d]
//   A = (Q K^T) / 8   (64x64), softmax rows, O = P V (64x256)
//   scatter O[d][w] -> mid[b][n*256 + d*4 + (w>>6)][h*64 + (w&63)]
// Q/K/V streamed through LDS in 64-wide w chunks; A kept in wave accumulators.
// ---------------------------------------------------------------------------
__global__ __launch_bounds__(256)
void attn_win(const float* __restrict__ qkv, float* __restrict__ mid)
{
    const int n = blockIdx.x;     // window 0..15
    const int h = blockIdx.y;     // head
    const int b = blockIdx.z;     // batch

    const int tid  = threadIdx.x;
    const int lane = tid & 31;
    const int wave = tid >> 5;
    const int r    = lane & 15;
    const int half = lane >> 4;

    __shared__ float Qs[64 * 64];        // current Q (phase1) / V (phase2) chunk, [w][d]
    __shared__ float Ks[64 * 64];        // current K chunk, [w][e]
    __shared__ float Ps[64 * 66];        // scaled A / softmax probs (stride 66: banks + 8B align)

    const float* base = qkv + (long long)b * (ROWS_Q * (long long)QKV_N)
                            + (long long)(n * 128) * QKV_N;
    const int qoff = 0 * EMBED + h * HEAD_D;
    const int koff = 1 * EMBED + h * HEAD_D;
    const int voff = 2 * EMBED + h * HEAD_D;

    const int ldw = tid >> 2;            // 0..63 : w row within chunk
    const int seg = (tid & 3) * 16;      // 16-float segment within the 64-float row

    // ---- phase 1: A = Q K^T accumulated over 4 w-chunks -------------------
    v8f accA0 = {}, accA1 = {};
    const int tile0 = wave * 2, tile1 = wave * 2 + 1;
    const int d0a = (tile0 >> 2) * 16, e0a = (tile0 & 3) * 16;
    const int d0b = (tile1 >> 2) * 16, e0b = (tile1 & 3) * 16;

    for (int wc = 0; wc < 4; ++wc) {
        const float* grow = base + (long long)(wc * 64 + ldw) * QKV_N;
#pragma unroll
        for (int j = 0; j < 16; j += 4) {
            *(float4*)&Qs[ldw * 64 + seg + j] = *(const float4*)&grow[qoff + seg + j];
            *(float4*)&Ks[ldw * 64 + seg + j] = *(const float4*)&grow[koff + seg + j];
        }
        __syncthreads();

#pragma unroll 4
        for (int k = 0; k < 64; k += 4) {
            const int k2 = k + 2 * half;
            v2f a, bb;
            a.x  = Qs[k2 * 64 + d0a + r];
            a.y  = Qs[(k2 + 1) * 64 + d0a + r];
            bb.x = Ks[k2 * 64 + e0a + r];
            bb.y = Ks[(k2 + 1) * 64 + e0a + r];
            accA0 = __builtin_amdgcn_wmma_f32_16x16x4_f32(
                false, a, false, bb, (short)0, accA0, false, false);
            a.x  = Qs[k2 * 64 + d0b + r];
            a.y  = Qs[(k2 + 1) * 64 + d0b + r];
            bb.x = Ks[k2 * 64 + e0b + r];
            bb.y = Ks[(k2 + 1) * 64 + e0b + r];
            accA1 = __builtin_amdgcn_wmma_f32_16x16x4_f32(
                false, a, false, bb, (short)0, accA1, false, false);
        }
        __syncthreads();
    }

    // ---- park scaled logits in LDS ---------------------------------------
#pragma unroll
    for (int i = 0; i < 8; ++i) {
        Ps[(d0a + half * 8 + i) * 66 + e0a + r] = accA0[i] * 0.125f;
        Ps[(d0b + half * 8 + i) * 66 + e0b + r] = accA1[i] * 0.125f;
    }
    __syncthreads();

    // ---- softmax over each of the 64 rows (64 lanes, padded stride) ------
    if (tid < 64) {
        float m = -3.402823466e38f;
        for (int e = 0; e < 64; ++e) m = fmaxf(m, Ps[tid * 66 + e]);
        float s = 0.f;
        for (int e = 0; e < 64; ++e) {
            float v = expf(Ps[tid * 66 + e] - m);
            Ps[tid * 66 + e] = v;
            s += v;
        }
        const float inv = 1.0f / s;
        for (int e = 0; e < 64; ++e) Ps[tid * 66 + e] *= inv;
    }
    __syncthreads();

    // ---- phase 2: O = P V per w-chunk, scatter to mid ---------------------
    float* midb = mid + (long long)b * SEQ * EMBED;
    for (int wc = 0; wc < 4; ++wc) {
        const float* grow = base + (long long)(wc * 64 + ldw) * QKV_N;
#pragma unroll
        for (int j = 0; j < 16; j += 4)
            *(float4*)&Qs[ldw * 64 + seg + j] = *(const float4*)&grow[voff + seg + j];
        __syncthreads();

#pragma unroll
        for (int t = 0; t < 2; ++t) {
            const int tile = wave * 2 + t;
            const int d0 = (tile >> 2) * 16;
            const int w0 = (tile & 3) * 16;
            v8f acc = {};
#pragma unroll 4
            for (int k = 0; k < 64; k += 4) {
                const int k2 = k + 2 * half;
                v2f a  = *(const v2f*)&Ps[(d0 + r) * 66 + k2];  // P[d][e..e+1]
                v2f bb = *(const v2f*)&Qs[(w0 + r) * 64 + k2];  // V[e..e+1][w]
                acc = __builtin_amdgcn_wmma_f32_16x16x4_f32(
                    false, a, false, bb, (short)0, acc, false, false);
            }
            const int c = w0 + r;                // (w & 63), since chunk-local w < 64
#pragma unroll
            for (int i = 0; i < 8; ++i) {
                const int d  = d0 + half * 8 + i;
                const int rr = n * 256 + d * 4 + wc;   // (w >> 6) == wc
                midb[(long long)rr * EMBED + h * HEAD_D + c] = acc[i];
            }
        }
        __syncthreads();
    }
}

// ---------------------------------------------------------------------------
extern "C" void kernel_launch(void* const* d_in, const int* in_sizes, int n_in,
                              void* d_out, int out_size, void* d_ws, size_t ws_size,
                              hipStream_t stream)
{
    const float* x      = (const float*)d_in[0];   // (4,4096,1024)
    const float* W_qkv  = (const float*)d_in[1];   // (1024,3072)
    const float* b_qkv  = (const float*)d_in[2];   // (3072,)
    const float* W_out  = (const float*)d_in[3];   // (1024,1024)
    const float* b_out  = (const float*)d_in[4];   // (1024,)
    float*       out    = (float*)d_out;           // (4,4096,1024)

    float* qkv_ws = (float*)d_ws;                              // 4*2176*3072 f32
    float* mid    = qkv_ws + (size_t)BATCH * ROWS_Q * QKV_N;   // 4*4096*1024 f32

    // 1) QKV projection, only the 2176 seq rows that feed windows 0..15
    gemm_bias_wmma<<<dim3(QKV_N / 256, ROWS_Q / 64, BATCH), 256, 0, stream>>>(
        x, W_qkv, b_qkv, qkv_ws,
        EMBED, QKV_N,
        (long long)SEQ * EMBED,            // x batch stride
        (long long)ROWS_Q * QKV_N);        // qkv batch stride

    // 2) windowed attention + scrambled scatter into mid
    attn_win<<<dim3(NW_USED, HEADS, BATCH), 256, 0, stream>>>(qkv_ws, mid);

    // 3) output projection over all B*4096 rows (mid is contiguous across batches)
    gemm_bias_wmma<<<dim3(EMBED / 256, (BATCH * SEQ) / 64, 1), 256, 0, stream>>>(
        mid, W_out, b_out, out,
        EMBED, EMBED, 0LL, 0LL);
}